// ResamplerLayer_38259568673124
// MI455X (gfx1250) — compile-verified
//
#include <hip/hip_runtime.h>

// Trilinear resampler, REPLICATE boundary, matching the JAX reference.
//   inputs:        [B=2, X=128, Y=128, Z=128, C=4] f32
//   sample_coords: [B=2, OX=96, OY=96, OZ=96, 3]   f32
//   out:           [B=2, OX=96, OY=96, OZ=96, C=4] f32
//
// Memory-bound gather kernel. MI455X reasoning:
//  - input volume (67 MB) fits in the 192 MB L2 -> gathers are L2 hits after
//    first touch; keep them at default (RT) temporal policy.
//  - coords + output are one-touch streams -> non-temporal (th:*_NT) so they
//    don't evict the resident volume.
//  - C=4 floats -> every corner gather is one aligned global_load_b128.

typedef float v4f __attribute__((ext_vector_type(4)));

namespace cfg {
constexpr int B  = 2;
constexpr int X  = 128, Y = 128, Z = 128, C = 4;
constexpr int OX = 96,  OY = 96, OZ = 96;
constexpr int PTS_PER_BATCH = OX * OY * OZ;          // 884736
constexpr int NPTS          = B * PTS_PER_BATCH;     // 1769472
constexpr int VOL_V4        = X * Y * Z;             // float4 units per batch
}

__global__ __launch_bounds__(256)
void resampler_trilinear_kernel(const float* __restrict__ in,
                                const float* __restrict__ coords,
                                float* __restrict__ out) {
    const int p = blockIdx.x * blockDim.x + threadIdx.x;
    if (p >= cfg::NPTS) return;

    // batch id without div/mod (B == 2)
    const int b     = (p >= cfg::PTS_PER_BATCH) ? 1 : 0;
    const int vbase = b * cfg::VOL_V4;

    // ---- streamed coord load (non-temporal: one-touch data) ----
    const size_t coff = (size_t)p * 3;
    const float fx = __builtin_nontemporal_load(coords + coff + 0);
    const float fy = __builtin_nontemporal_load(coords + coff + 1);
    const float fz = __builtin_nontemporal_load(coords + coff + 2);

    // ---- floor / frac / REPLICATE clamp ----
    const float bx = floorf(fx), by = floorf(fy), bz = floorf(fz);
    const float wx1 = fx - bx, wy1 = fy - by, wz1 = fz - bz;   // weight for ceil
    const float wx0 = 1.0f - wx1, wy0 = 1.0f - wy1, wz0 = 1.0f - wz1;

    const int ix0 = (int)fminf(fmaxf(bx,        0.0f), (float)(cfg::X - 1));
    const int ix1 = (int)fminf(fmaxf(bx + 1.0f, 0.0f), (float)(cfg::X - 1));
    const int iy0 = (int)fminf(fmaxf(by,        0.0f), (float)(cfg::Y - 1));
    const int iy1 = (int)fminf(fmaxf(by + 1.0f, 0.0f), (float)(cfg::Y - 1));
    const int iz0 = (int)fminf(fmaxf(bz,        0.0f), (float)(cfg::Z - 1));
    const int iz1 = (int)fminf(fmaxf(bz + 1.0f, 0.0f), (float)(cfg::Z - 1));

    // float4-unit offsets (X=Y=Z=128 -> shifts)
    const int x0 = ix0 << 14, x1 = ix1 << 14;   // * Y*Z
    const int y0 = iy0 << 7,  y1 = iy1 << 7;    // * Z

    const v4f* __restrict__ in4 = reinterpret_cast<const v4f*>(in);

    // ---- issue all 8 b128 gathers up front (memory-level parallelism) ----
    const v4f c000 = in4[vbase + x0 + y0 + iz0];
    const v4f c001 = in4[vbase + x0 + y0 + iz1];
    const v4f c010 = in4[vbase + x0 + y1 + iz0];
    const v4f c011 = in4[vbase + x0 + y1 + iz1];
    const v4f c100 = in4[vbase + x1 + y0 + iz0];
    const v4f c101 = in4[vbase + x1 + y0 + iz1];
    const v4f c110 = in4[vbase + x1 + y1 + iz0];
    const v4f c111 = in4[vbase + x1 + y1 + iz1];

    // ---- trilinear blend (partial products: 4 + 8 mults) ----
    const float wxy00 = wx0 * wy0;
    const float wxy01 = wx0 * wy1;
    const float wxy10 = wx1 * wy0;
    const float wxy11 = wx1 * wy1;

    v4f acc = c000 * (wxy00 * wz0);
    acc += c001 * (wxy00 * wz1);
    acc += c010 * (wxy01 * wz0);
    acc += c011 * (wxy01 * wz1);
    acc += c100 * (wxy10 * wz0);
    acc += c101 * (wxy10 * wz1);
    acc += c110 * (wxy11 * wz0);
    acc += c111 * (wxy11 * wz1);

    // ---- streamed output store (non-temporal b128) ----
    v4f* __restrict__ out4 = reinterpret_cast<v4f*>(out);
    __builtin_nontemporal_store(acc, out4 + p);
}

extern "C" void kernel_launch(void* const* d_in, const int* in_sizes, int n_in,
                              void* d_out, int out_size, void* d_ws, size_t ws_size,
                              hipStream_t stream) {
    const float* inputs = (const float*)d_in[0];   // [2,128,128,128,4] f32
    const float* coords = (const float*)d_in[1];   // [2,96,96,96,3]    f32
    float*       out    = (float*)d_out;           // [2,96,96,96,4]    f32

    constexpr int threads = 256;                   // 8 wave32's
    const int blocks = (cfg::NPTS + threads - 1) / threads;  // 6912
    resampler_trilinear_kernel<<<blocks, threads, 0, stream>>>(inputs, coords, out);
}